// DLSTM_24962349925126
// MI455X (gfx1250) — compile-verified
//
#include <hip/hip_runtime.h>

// DLSTM recurrent scan on gfx1250 using V_WMMA_F32_16X16X4_F32 (full fp32
// matrix cores; preserves reference precision over the 512-step recurrence).
//
// Per timestep:
//   kernel 1: h1 = relu([x_t | c] @ W0^T + b0)         (128x512, K=1024)
//   kernel 2: net = h1 @ W1^T + b1 (5 gate blocks fused per tile),
//             c    = sigmoid(i)*drelu(a,b) + sigmoid(f)*c
//             out_t = relu(net[:,2H:3H])
//
// Weights (7 MB) are L2-resident (192 MB L2); the scan is launch/latency
// bound, so fp32 WMMA costs ~nothing vs bf16 while matching fp32 numerics.

typedef float v2f __attribute__((ext_vector_type(2)));
typedef float v8f __attribute__((ext_vector_type(8)));

#define T_STEPS 512
#define BATCH   128
#define INSZ    512
#define HID     512

__device__ __forceinline__ v2f ldf2(const float* p) {
    return *reinterpret_cast<const v2f*>(p);
}

__device__ __forceinline__ v8f wmma_f32(v2f a, v2f b, v8f c) {
    return __builtin_amdgcn_wmma_f32_16x16x4_f32(false, a, false, b,
                                                 (short)0, c, false, false);
}

// ---------------------------------------------------------------------------
// GEMM1: h1[128 x 512] = relu( [x_t | c] (128x1024) @ W0^T (1024x512) + b0 )
// One wave per workgroup; each wave owns a 32(M) x 16(N) output block
// (two 16x16 WMMA tiles sharing the B fragment).
// grid = (512/16 = 32 Ntiles, 128/32 = 4 Mblocks), block = 32 (wave32)
// ---------------------------------------------------------------------------
__global__ __launch_bounds__(32) void dlstm_gemm1(
    const float* __restrict__ x,   // (T, B, I)
    const float* __restrict__ c,   // (B, H)  current cell state
    const float* __restrict__ W0,  // (H, H+I) row-major
    const float* __restrict__ b0,  // (H)
    float* __restrict__ h1,        // (B, H)
    int t)
{
    const int lane = threadIdx.x;
    const int lo = lane & 15;      // A: row M   | B: col N   | C: col N
    const int hi = lane >> 4;      // K-pair select (K = 2*hi + {0,1})
    const int ks = 2 * hi;
    const int nt = blockIdx.x * 16;
    const int mt = blockIdx.y * 32;

    const float* xrow0 = x + ((size_t)t * BATCH + (mt + lo)) * INSZ;
    const float* xrow1 = xrow0 + (size_t)16 * INSZ;
    const float* crow0 = c + (size_t)(mt + lo) * HID;
    const float* crow1 = crow0 + (size_t)16 * HID;
    // B[k][n] = W0[nt+lo][k] -> contiguous float2 along k
    const float* wrow = W0 + (size_t)(nt + lo) * (INSZ + HID);

    v8f acc0 = {};
    v8f acc1 = {};

    // K in [0, 512): A comes from x_t
    #pragma unroll 8
    for (int k = 0; k < INSZ; k += 4) {
        v2f a0 = ldf2(xrow0 + k + ks);
        v2f a1 = ldf2(xrow1 + k + ks);
        v2f b  = ldf2(wrow  + k + ks);
        acc0 = wmma_f32(a0, b, acc0);
        acc1 = wmma_f32(a1, b, acc1);
    }
    // K in [512, 1024): A comes from c
    #pragma unroll 8
    for (int k = 0; k < HID; k += 4) {
        v2f a0 = ldf2(crow0 + k + ks);
        v2f a1 = ldf2(crow1 + k + ks);
        v2f b  = ldf2(wrow + INSZ + k + ks);
        acc0 = wmma_f32(a0, b, acc0);
        acc1 = wmma_f32(a1, b, acc1);
    }

    const float bias = b0[nt + lo];
    #pragma unroll
    for (int v = 0; v < 8; ++v) {
        const int r = v + 8 * hi;                 // C/D row within tile
        float u0 = acc0[v] + bias;
        float u1 = acc1[v] + bias;
        u0 = u0 > 0.0f ? u0 : 0.0f;
        u1 = u1 > 0.0f ? u1 : 0.0f;
        h1[(size_t)(mt + r) * HID + (nt + lo)]      = u0;
        h1[(size_t)(mt + 16 + r) * HID + (nt + lo)] = u1;
    }
}

// ---------------------------------------------------------------------------
// GEMM2 + gates: for a 16(M) x 16(N) tile of gate-columns, compute all 5
// gate blocks net[:, g*H + n] (5 accumulators share one A fragment), then
// the full LSTM-style epilogue in-register. In-place c update is race-free:
// each c element is owned by exactly one lane of one wave.
// grid = (512/16 = 32 Ntiles, 128/16 = 8 Mtiles), block = 32 (wave32)
// ---------------------------------------------------------------------------
__global__ __launch_bounds__(32) void dlstm_gemm2(
    const float* __restrict__ h1,  // (B, H)
    const float* __restrict__ W1,  // (5H, H) row-major
    const float* __restrict__ b1,  // (5H)
    float* __restrict__ c,         // (B, H) updated in place
    float* __restrict__ out,       // (T, B, H)
    int t)
{
    const int lane = threadIdx.x;
    const int lo = lane & 15;
    const int hi = lane >> 4;
    const int ks = 2 * hi;
    const int nt = blockIdx.x * 16;
    const int mt = blockIdx.y * 16;

    const float* arow = h1 + (size_t)(mt + lo) * HID;
    const float* w0r = W1 + (size_t)(0 * HID + nt + lo) * HID;
    const float* w1r = W1 + (size_t)(1 * HID + nt + lo) * HID;
    const float* w2r = W1 + (size_t)(2 * HID + nt + lo) * HID;
    const float* w3r = W1 + (size_t)(3 * HID + nt + lo) * HID;
    const float* w4r = W1 + (size_t)(4 * HID + nt + lo) * HID;

    v8f acc0 = {}, acc1 = {}, acc2 = {}, acc3 = {}, acc4 = {};

    #pragma unroll 4
    for (int k = 0; k < HID; k += 4) {
        v2f a = ldf2(arow + k + ks);
        acc0 = wmma_f32(a, ldf2(w0r + k + ks), acc0);
        acc1 = wmma_f32(a, ldf2(w1r + k + ks), acc1);
        acc2 = wmma_f32(a, ldf2(w2r + k + ks), acc2);
        acc3 = wmma_f32(a, ldf2(w3r + k + ks), acc3);
        acc4 = wmma_f32(a, ldf2(w4r + k + ks), acc4);
    }

    const int col = nt + lo;
    const float bb0 = b1[0 * HID + col];
    const float bb1 = b1[1 * HID + col];
    const float bb2 = b1[2 * HID + col];
    const float bb3 = b1[3 * HID + col];
    const float bb4 = b1[4 * HID + col];

    #pragma unroll
    for (int v = 0; v < 8; ++v) {
        const int row = mt + v + 8 * hi;
        const float n0 = acc0[v] + bb0;           // cx_a
        const float n1 = acc1[v] + bb1;           // cx_b
        const float n2 = acc2[v] + bb2;           // new_hx pre-relu
        const float n3 = acc3[v] + bb3;           // forget gate pre-sigmoid
        const float n4 = acc4[v] + bb4;           // input gate pre-sigmoid

        const float hx = fmaxf(n2, 0.0f);
        const float cx = fmaxf(n0, 0.0f) - fmaxf(n1, 0.0f);
        const float fg = 1.0f / (1.0f + __expf(-n3));
        const float ig = 1.0f / (1.0f + __expf(-n4));

        const size_t ci = (size_t)row * HID + col;
        const float cnew = ig * cx + fg * c[ci];
        c[ci] = cnew;
        out[((size_t)t * BATCH + row) * HID + col] = hx;
    }
}

// ---------------------------------------------------------------------------
// Host-side: 512 steps x 2 kernels, all on `stream` (graph-capture safe).
// ---------------------------------------------------------------------------
extern "C" void kernel_launch(void* const* d_in, const int* in_sizes, int n_in,
                              void* d_out, int out_size, void* d_ws, size_t ws_size,
                              hipStream_t stream) {
    const float* x  = (const float*)d_in[0];  // (T, B, I)
    const float* c0 = (const float*)d_in[1];  // (1, B, H)
    const float* W0 = (const float*)d_in[2];  // (H, H+I)
    const float* b0 = (const float*)d_in[3];  // (H)
    const float* W1 = (const float*)d_in[4];  // (5H, H)
    const float* b1 = (const float*)d_in[5];  // (5H)
    float* out = (float*)d_out;               // T*B*H outputs, then B*H c_final

    float* c_ws  = (float*)d_ws;                      // (B, H)
    float* h1_ws = c_ws + (size_t)BATCH * HID;        // (B, H)

    // Initialize carried cell state from the (immutable) input every call.
    hipMemcpyAsync(c_ws, c0, (size_t)BATCH * HID * sizeof(float),
                   hipMemcpyDeviceToDevice, stream);

    const dim3 blk(32, 1, 1);
    const dim3 grid1(HID / 16, BATCH / 32, 1);   // 32 x 4
    const dim3 grid2(HID / 16, BATCH / 16, 1);   // 32 x 8

    for (int t = 0; t < T_STEPS; ++t) {
        dlstm_gemm1<<<grid1, blk, 0, stream>>>(x, c_ws, W0, b0, h1_ws, t);
        dlstm_gemm2<<<grid2, blk, 0, stream>>>(h1_ws, W1, b1, c_ws, out, t);
    }

    // c_final goes after the (T, B, H) outputs, flat in return order.
    hipMemcpyAsync(out + (size_t)T_STEPS * BATCH * HID, c_ws,
                   (size_t)BATCH * HID * sizeof(float),
                   hipMemcpyDeviceToDevice, stream);
}